// Model_36730560315490
// MI455X (gfx1250) — compile-verified
//
#include <hip/hip_runtime.h>
#include <hip/hip_bf16.h>

// Reference structure: B=1M independent 2-var LM problems, but residual() does
// NOT depend on y, and x0 == 0 for every row => all rows are identical.
// Strategy: (1) one wave runs the 50 serial LM iterations exactly once,
// using V_WMMA_F32_16X16X4_F32 for the 16x2 hidden-layer matvec (H=16 matches
// the WMMA tile), (2) broadcast the final 2-vector to (B,2) with B128 stores
// (8 MB -> ~0.34 us at 23.3 TB/s HBM).

typedef __attribute__((ext_vector_type(2))) float v2f;
typedef __attribute__((ext_vector_type(8))) float v8f;

#define MAX_ITERS 50
#define STEP_SIZE 1e-4f

// ---------------------------------------------------------------------------
// Kernel 1: single-wave LM solver (1 block x 32 threads, EXEC all-ones so
// WMMA is legal). A-matrix (16x4 f32) layout per ISA: VGPR0 = K0 (lanes 0-15)
// / K2 (lanes 16-31), VGPR1 = K1 / K3. We zero K=2,3 so the B operand is
// correct under either K->(vgpr,half) convention.
// ---------------------------------------------------------------------------
__global__ __launch_bounds__(32) void lm_solve_kernel(
    const float* __restrict__ W1,   // (16,2) row-major
    const float* __restrict__ b1,   // (16,)
    const float* __restrict__ W2,   // (1,16)
    const float* __restrict__ b2,   // (1,)
    float* __restrict__ sol)        // out: 2 floats
{
  const int  lane = threadIdx.x;      // 0..31
  const bool lo   = lane < 16;
  const int  mb   = lo ? 0 : 8;       // which 8 rows of the 16 this lane owns in C/D

  // A = W1 padded to 16x4 (columns K=2,3 are zero)
  v2f amat;
  amat.x = lo ? W1[lane * 2 + 0] : 0.0f;   // A[M=lane][K=0]   (K=2 half -> 0)
  amat.y = lo ? W1[lane * 2 + 1] : 0.0f;   // A[M=lane][K=1]   (K=3 half -> 0)

  // C = b1 replicated across all 16 columns: c[v] = b1[mb+v] in every lane.
  v8f cbias;
#pragma unroll
  for (int v = 0; v < 8; ++v) cbias[v] = b1[mb + v];

  // Per-lane copies of the 8 rows this lane owns, for the epilogue reductions.
  float w2l[8], w1k0[8], w1k1[8];
#pragma unroll
  for (int v = 0; v < 8; ++v) {
    w2l[v]  = W2[mb + v];
    w1k0[v] = W1[(mb + v) * 2 + 0];
    w1k1[v] = W1[(mb + v) * 2 + 1];
  }
  const float bias2 = b2[0];

  float x0 = 0.0f, x1 = 0.0f;

  for (int it = 0; it < MAX_ITERS; ++it) {
    // clamp + clip-derivative mask
    const float xc0 = fminf(fmaxf(x0, -1.0f), 1.0f);
    const float xc1 = fminf(fmaxf(x1, -1.0f), 1.0f);
    const float m0  = (x0 > -1.0f && x0 < 1.0f) ? 1.0f : 0.0f;
    const float m1  = (x1 > -1.0f && x1 < 1.0f) ? 1.0f : 0.0f;

    // B (4x16): every column = (xc0, xc1, 0, 0)^T. Putting xc1 in the upper
    // half of VGPR0 makes this correct whether that slot is row K=1 or K=2
    // (A's K=2 column is zero either way).
    v2f bmat;
    bmat.x = lo ? xc0 : xc1;
    bmat.y = lo ? xc1 : 0.0f;

    // D = A*B + C : every lane holds pre[mb..mb+7] (all 16 columns identical)
    v8f d = __builtin_amdgcn_wmma_f32_16x16x4_f32(
        false, amat, false, bmat, (short)0, cbias, false, false);

    // Epilogue partials over this lane's 8 hidden units
    float s_out = 0.0f, sj0 = 0.0f, sj1 = 0.0f;
#pragma unroll
    for (int v = 0; v < 8; ++v) {
      const float pre   = d[v];
      const float act   = pre > 0.0f ? pre : 0.0f;   // relu
      const float gmask = pre > 0.0f ? 1.0f : 0.0f;  // relu'
      s_out = fmaf(w2l[v], act, s_out);
      sj0   = fmaf(w2l[v] * gmask, w1k0[v], sj0);
      sj1   = fmaf(w2l[v] * gmask, w1k1[v], sj1);
    }
    // combine the two lane-halves (wave32 xor-16 exchange)
    s_out += __shfl_xor(s_out, 16, 32);
    sj0   += __shfl_xor(sj0,   16, 32);
    sj1   += __shfl_xor(sj1,   16, 32);

    const float r  = 1000.0f - (s_out + bias2);      // residual (scalar)
    const float J0 = -sj0 * m0;                      // dr/dx0
    const float J1 = -sj1 * m1;                      // dr/dx1

    // Damped normal equations: (J^T J + I) dx = -J^T r   (closed-form 2x2)
    const float g0  = -J0 * r;
    const float g1  = -J1 * r;
    const float a11 = fmaf(J0, J0, 1.0f);
    const float a22 = fmaf(J1, J1, 1.0f);
    const float a12 = J0 * J1;
    const float inv = 1.0f / (a11 * a22 - a12 * a12);
    const float dx0 = (a22 * g0 - a12 * g1) * inv;
    const float dx1 = (a11 * g1 - a12 * g0) * inv;

    x0 = fmaf(STEP_SIZE, dx0, x0);
    x1 = fmaf(STEP_SIZE, dx1, x1);
  }

  if (lane == 0) {
    sol[0] = x0;
    sol[1] = x1;
  }
}

// ---------------------------------------------------------------------------
// Kernel 2: broadcast (x0,x1) to all B rows with 128-bit stores.
// ---------------------------------------------------------------------------
__global__ __launch_bounds__(256) void bcast_kernel(
    const float* __restrict__ sol, float* __restrict__ out,
    int nvec4, int rem)
{
  const int i  = blockIdx.x * blockDim.x + threadIdx.x;
  const float x0 = sol[0];
  const float x1 = sol[1];
  if (i < nvec4) {
    float4 v;
    v.x = x0; v.y = x1; v.z = x0; v.w = x1;
    reinterpret_cast<float4*>(out)[i] = v;   // global_store_b128
  }
  if (i == 0) {
    for (int t = 0; t < rem; ++t)            // out_size % 4 tail (0 or 2)
      out[nvec4 * 4 + t] = (t & 1) ? x1 : x0;
  }
}

extern "C" void kernel_launch(void* const* d_in, const int* in_sizes, int n_in,
                              void* d_out, int out_size, void* d_ws, size_t ws_size,
                              hipStream_t stream) {
  // setup_inputs order: y (unused), W1, b1, W2, b2
  const float* W1 = (const float*)d_in[1];
  const float* b1 = (const float*)d_in[2];
  const float* W2 = (const float*)d_in[3];
  const float* b2 = (const float*)d_in[4];
  float* out = (float*)d_out;
  float* sol = (float*)d_ws;   // 2 floats of scratch

  lm_solve_kernel<<<1, 32, 0, stream>>>(W1, b1, W2, b2, sol);

  const int nvec4  = out_size >> 2;
  const int rem    = out_size & 3;
  const int blocks = (nvec4 + 255) / 256 > 0 ? (nvec4 + 255) / 256 : 1;
  bcast_kernel<<<blocks, 256, 0, stream>>>(sol, out, nvec4, rem);
}